// ROIBoxHead_17162689315290
// MI455X (gfx1250) — compile-verified
//
#include <hip/hip_runtime.h>

#define BATCH 32
#define NPROP 2048
#define NCLS  601
#define CP    608     // NCLS padded to 38*16
#define GBLK  8       // partial-max blocks per image
#define ITERS 10

typedef __attribute__((ext_vector_type(2))) float v2f;
typedef __attribute__((ext_vector_type(8))) float v8f;

// ---------------------------------------------------------------------------
// ws layout (floats): [0, BATCH*CP)                      pw  (w * msg @ CM)
//                     [BATCH*CP, BATCH*CP+BATCH*GBLK*CP) partial maxes
// ---------------------------------------------------------------------------

__global__ void crf_init(float* __restrict__ pw) {
    int i = blockIdx.x * blockDim.x + threadIdx.x;
    if (i < BATCH * CP) pw[i] = 0.0f;
}

// One block = one (image b, row-group g): 8 waves x 32 rows = 256 rows.
// Lane l owns columns {l, l+32, ...} (19 per lane). Wave32 shuffles for the
// row max / row sum; block combine via LDS uint atomicMax (values > 0).
__global__ __launch_bounds__(256)
void crf_softmax_max(const float* __restrict__ u, const float* __restrict__ pw,
                     float* __restrict__ partial) {
    __shared__ unsigned int smax[CP];
    const int tid  = threadIdx.x;
    const int lane = tid & 31;
    const int wave = tid >> 5;
    const int b    = blockIdx.x >> 3;
    const int g    = blockIdx.x & 7;

    for (int i = tid; i < CP; i += 256) smax[i] = 0u;
    __syncthreads();

    float pwv[19];
#pragma unroll
    for (int j = 0; j < 19; ++j) {
        int c = lane + 32 * j;
        pwv[j] = (c < NCLS) ? pw[b * CP + c] : 0.0f;
    }
    float lmax[19];
#pragma unroll
    for (int j = 0; j < 19; ++j) lmax[j] = 0.0f;

    const int row0 = g * 256 + wave * 32;
    for (int i = 0; i < 32; ++i) {
        const float* up = u + (size_t)(b * NPROP + row0 + i) * NCLS;
        float v[19];
        float m = -3.4e38f;
#pragma unroll
        for (int j = 0; j < 19; ++j) {
            int c = lane + 32 * j;
            if (c < NCLS) { v[j] = up[c] - pwv[j]; m = fmaxf(m, v[j]); }
            else          { v[j] = -3.4e38f; }
        }
#pragma unroll
        for (int off = 16; off > 0; off >>= 1) m = fmaxf(m, __shfl_xor(m, off, 32));
        float s = 0.0f;
#pragma unroll
        for (int j = 0; j < 19; ++j) {
            int c = lane + 32 * j;
            if (c < NCLS) { float e = __expf(v[j] - m); v[j] = e; s += e; }
        }
#pragma unroll
        for (int off = 16; off > 0; off >>= 1) s += __shfl_xor(s, off, 32);
        float inv = 1.0f / s;
#pragma unroll
        for (int j = 0; j < 19; ++j) {
            int c = lane + 32 * j;
            if (c < NCLS) lmax[j] = fmaxf(lmax[j], v[j] * inv);
        }
    }
#pragma unroll
    for (int j = 0; j < 19; ++j) {
        int c = lane + 32 * j;
        if (c < NCLS) atomicMax(&smax[c], __float_as_uint(lmax[j]));
    }
    __syncthreads();
    // write all CP entries (padding = 0) so downstream needs no guards
    float* prow = partial + (size_t)(b * GBLK + g) * CP;
    for (int c = tid; c < CP; c += 256) prow[c] = __uint_as_float(smax[c]);
}

// grid (38, 2): blockIdx.x = n-tile, blockIdx.y = m-tile (16 images each).
// Phase 1: reduce 8 partials -> msgLDS[16][608] (stride 609 => bank-conflict
// free A-fragment reads); padding columns are zero.
// Phase 2: wave 0 accumulates the 16x16 tile with v_wmma_f32_16x16x4_f32.
// Branchless B loads: A is zero for padded K, so clamped (in-bounds) B
// addresses are sufficient; out-of-range N columns land in pw's padding.
__global__ __launch_bounds__(256)
void crf_msg_gemm(const float* __restrict__ partial, const float* __restrict__ cm,
                  const float* __restrict__ wptr, float* __restrict__ pw) {
    __shared__ float msgLDS[16][CP + 1];
    const int tid = threadIdx.x;
    const int b0  = blockIdx.y * 16;

    for (int idx = tid; idx < 16 * CP; idx += 256) {
        int bi = idx / CP, c = idx - bi * CP;
        float mx = 0.0f;
#pragma unroll
        for (int g = 0; g < GBLK; ++g)
            mx = fmaxf(mx, partial[(size_t)((b0 + bi) * GBLK + g) * CP + c]);
        msgLDS[bi][c] = mx;
    }
    __syncthreads();

    if (tid < 32) {
        const int lane = tid;
        const int r    = lane & 15;            // row (A) / col (B,D) within tile
        const int koff = (lane >> 4) << 1;     // lanes 16..31 carry K+2, K+3
        const int n0   = blockIdx.x * 16;
        const int ncol = n0 + r;
        const float w  = wptr[0];
        // clamp column pointer once; garbage columns only reach pw padding
        const float* cmcol = cm + (ncol < NCLS ? ncol : (NCLS - 1));

        v8f acc = {0.f, 0.f, 0.f, 0.f, 0.f, 0.f, 0.f, 0.f};
        int k0 = 0;
        // main loop: k0+3 <= 599 < NCLS, all row indices valid -> no clamps
#pragma unroll 4
        for (; k0 < 600; k0 += 4) {
            const int ka = k0 + koff;
            v2f a, bb;
            a.x  = msgLDS[r][ka];
            a.y  = msgLDS[r][ka + 1];
            bb.x = cmcol[(size_t)ka * NCLS];
            bb.y = cmcol[(size_t)(ka + 1) * NCLS];
            acc = __builtin_amdgcn_wmma_f32_16x16x4_f32(
                false, a, false, bb, (short)0, acc, false, false);
        }
        // tail: clamp row indices (A is zero for k >= NCLS, so values don't matter)
        for (; k0 < CP; k0 += 4) {
            const int ka = k0 + koff;
            const int r0 = (ka     < NCLS) ? ka     : (NCLS - 1);
            const int r1 = (ka + 1 < NCLS) ? ka + 1 : (NCLS - 1);
            v2f a, bb;
            a.x  = msgLDS[r][ka];
            a.y  = msgLDS[r][ka + 1];
            bb.x = cmcol[(size_t)r0 * NCLS];
            bb.y = cmcol[(size_t)r1 * NCLS];
            acc = __builtin_amdgcn_wmma_f32_16x16x4_f32(
                false, a, false, bb, (short)0, acc, false, false);
        }
        // D layout: VGPR g -> M = g (lanes 0-15) / M = 8+g (lanes 16-31), N = lane&15
#pragma unroll
        for (int vg = 0; vg < 8; ++vg) {
            int m = b0 + vg + ((lane >> 4) << 3);
            pw[m * CP + ncol] = w * acc[vg];
        }
    }
}

__global__ __launch_bounds__(256)
void crf_finalize(const float* __restrict__ u, const float* __restrict__ pw,
                  float* __restrict__ out) {
    const int row  = blockIdx.x;
    const int b    = row >> 11;             // NPROP == 2048
    const size_t base = (size_t)row * NCLS;
    const float* pwb = pw + b * CP;
    for (int c = threadIdx.x; c < NCLS; c += 256)
        out[base + c] = u[base + c] - pwb[c];
}

extern "C" void kernel_launch(void* const* d_in, const int* in_sizes, int n_in,
                              void* d_out, int out_size, void* d_ws, size_t ws_size,
                              hipStream_t stream) {
    const float* u  = (const float*)d_in[0];   // [32,2048,601] f32
    const float* cm = (const float*)d_in[1];   // [601,601]     f32
    const float* w  = (const float*)d_in[2];   // [1]           f32
    float* out      = (float*)d_out;

    float* pw      = (float*)d_ws;             // [32][608]
    float* partial = pw + BATCH * CP;          // [32][8][608]

    crf_init<<<(BATCH * CP + 255) / 256, 256, 0, stream>>>(pw);

    dim3 ggrid(CP / 16, 2);
    for (int it = 0; it < ITERS; ++it) {
        crf_softmax_max<<<BATCH * GBLK, 256, 0, stream>>>(u, pw, partial);
        crf_msg_gemm<<<ggrid, 256, 0, stream>>>(partial, cm, w, pw);
    }
    crf_finalize<<<BATCH * NPROP, 256, 0, stream>>>(u, pw, out);
}